// simplicial_conv_in_71485435675276
// MI455X (gfx1250) — compile-verified
//
#include <hip/hip_runtime.h>
#include <cmath>

// ---------------------------------------------------------------------------
// Problem constants (match reference)
// ---------------------------------------------------------------------------
#define N_SIMP 8192   // Laplacian dimension N
#define N_FILT 8      // K filters
#define BATCH  64     // B
#define FEAT   128    // F

// K-split for occupancy: 8 chunks of 1024 reduction steps each
#define KCHUNK  1024
#define NCHUNKS (N_SIMP / KCHUNK)   // 8

// Workspace layout (in floats):
//   Wpad         : [16][N_SIMP]               (W with rows 8..15 zeroed)
//   ws1 partials : [NCHUNKS][16][N_SIMP]      (stage-1 partial wL, padded M=16)
//   wLpad        : [16][N_SIMP]               (reduced, rows 8..15 are zero)
//   ws2 partials : [NCHUNKS][N_FILT][BATCH][FEAT]
#define WPAD_OFF ((size_t)0)
#define WS1_OFF  ((size_t)16 * N_SIMP)
#define WLP_OFF  (WS1_OFF + (size_t)NCHUNKS * 16 * N_SIMP)
#define WS2_OFF  (WLP_OFF + (size_t)16 * N_SIMP)
// total = (131072 + 1048576 + 131072 + 524288) floats ~= 7.34 MB

typedef __attribute__((ext_vector_type(2))) float v2f;
typedef __attribute__((ext_vector_type(8))) float v8f;

// ---------------------------------------------------------------------------
// Prologue: Wpad[16][N] = { W rows 0..7 ; zeros rows 8..15 }.
// 256 KB copy + 256 KB zero fill — negligible; keeps the stage-1 inner loop
// free of per-iteration lane masking (no v_cndmask before each WMMA).
// ---------------------------------------------------------------------------
__global__ __launch_bounds__(256)
void k_pad_W(const float* __restrict__ W, float* __restrict__ ws) {
  const size_t i = (size_t)blockIdx.x * 256 + threadIdx.x;  // 16*8192 = 131072
  const size_t row = i / N_SIMP;
  ws[WPAD_OFF + i] = (row < N_FILT) ? W[i] : 0.f;
}

// ---------------------------------------------------------------------------
// Stage 1: partial wL = Wpad(16 rows) @ L over one K-chunk.
// One wave handles one 16-column N-tile for one K-chunk.
// A fragment (16x4 f32): lane l -> row m=l%16, K pair {k0+2h, k0+2h+1}, h=l/16.
// B fragment (4x16 f32): lane l -> col m=l%16, K pair {k0+2h, k0+2h+1}.
// C/D (16x16 f32): VGPR r, lane l -> M = r + 8h, N = m.
// ---------------------------------------------------------------------------
__global__ __launch_bounds__(128)
void k_wL_partial(const float* __restrict__ L, float* __restrict__ ws) {
  const int lane = threadIdx.x & 31;
  const int wid  = blockIdx.x * (blockDim.x >> 5) + (threadIdx.x >> 5);
  const int nt   = wid & 511;   // 512 N-tiles of 16 columns
  const int kc   = wid >> 9;    // 8 K-chunks
  const int m    = lane & 15;
  const int h    = lane >> 4;

  const int col = nt * 16 + m;
  const float* __restrict__ Lp = L + (size_t)(kc * KCHUNK + 2 * h) * N_SIMP + col;
  const float* __restrict__ Ap =
      ws + WPAD_OFF + (size_t)m * N_SIMP + kc * KCHUNK + 2 * h;

  v8f acc = {0.f, 0.f, 0.f, 0.f, 0.f, 0.f, 0.f, 0.f};

  #pragma unroll 4
  for (int s = 0; s < KCHUNK / 4; ++s) {
    if ((s & 7) == 0)  // prime the streamed-B pipe ~32 rows ahead
      __builtin_prefetch(Lp + (size_t)32 * N_SIMP, 0, 1);

    v2f a = *(const v2f*)Ap;   // rows 8..15 hold exact zeros (Wpad)
    v2f b;
    b.x = Lp[0];
    b.y = Lp[N_SIMP];
    acc = __builtin_amdgcn_wmma_f32_16x16x4_f32(
        /*neg_a=*/false, a, /*neg_b=*/false, b,
        /*c_mod=*/(short)0, acc, /*reuse_a=*/false, /*reuse_b=*/false);

    Lp += (size_t)4 * N_SIMP;
    Ap += 4;
  }

  // store full padded 16xN tile (rows 8..15 are exact zeros)
  float* __restrict__ o =
      ws + WS1_OFF + (size_t)kc * 16 * N_SIMP + (size_t)(8 * h) * N_SIMP + col;
  #pragma unroll
  for (int r = 0; r < 8; ++r)
    o[(size_t)r * N_SIMP] = acc[r];
}

// ---------------------------------------------------------------------------
// Stage 1 reduce: wLpad[16][N] = sum over NCHUNKS partials (deterministic).
// ---------------------------------------------------------------------------
__global__ __launch_bounds__(256)
void k_wL_reduce(float* __restrict__ ws) {
  const size_t i = (size_t)blockIdx.x * 256 + threadIdx.x;  // 16*8192 = 131072
  float s = 0.f;
  #pragma unroll
  for (int c = 0; c < NCHUNKS; ++c)
    s += ws[WS1_OFF + (size_t)c * 16 * N_SIMP + i];
  ws[WLP_OFF + i] = s;
}

// ---------------------------------------------------------------------------
// Stage 2: partial out = wLpad(16xN) @ x[b] (N x FEAT) over one N-chunk,
// per (batch, 16-wide feature tile). A rows 8..15 are zero -> C rows 8..15
// are zero and lanes 16..31 simply don't store.
// ---------------------------------------------------------------------------
__global__ __launch_bounds__(128)
void k_out_partial(const float* __restrict__ x, float* __restrict__ ws) {
  const int lane = threadIdx.x & 31;
  const int wid  = blockIdx.x * (blockDim.x >> 5) + (threadIdx.x >> 5);
  // jobs: nc(8) x b(64) x ft(8) = 4096 waves
  const int ft = wid & 7;
  const int b  = (wid >> 3) & 63;
  const int nc = wid >> 9;
  const int m  = lane & 15;
  const int h  = lane >> 4;

  const float* __restrict__ Ap =
      ws + WLP_OFF + (size_t)m * N_SIMP + nc * KCHUNK + 2 * h;
  const float* __restrict__ Bp = x + (size_t)b * N_SIMP * FEAT
                                   + (size_t)(nc * KCHUNK + 2 * h) * FEAT
                                   + ft * 16 + m;

  v8f acc = {0.f, 0.f, 0.f, 0.f, 0.f, 0.f, 0.f, 0.f};

  #pragma unroll 4
  for (int s = 0; s < KCHUNK / 4; ++s) {
    if ((s & 7) == 0)  // prefetch the streamed-x pipe ahead
      __builtin_prefetch(Bp + (size_t)32 * FEAT, 0, 1);

    v2f a = *(const v2f*)Ap;   // padded rows 8..15 hold exact zeros
    v2f bv;
    bv.x = Bp[0];
    bv.y = Bp[FEAT];
    acc = __builtin_amdgcn_wmma_f32_16x16x4_f32(
        false, a, false, bv, (short)0, acc, false, false);

    Ap += 4;
    Bp += (size_t)4 * FEAT;
  }

  // rows M = r + 8h; only M < 8 (h == 0) are real filter outputs
  if (h == 0) {
    float* __restrict__ o = ws + WS2_OFF
        + (size_t)nc * (N_FILT * BATCH * FEAT)
        + (size_t)b * FEAT + ft * 16 + m;
    #pragma unroll
    for (int r = 0; r < 8; ++r)
      o[(size_t)r * (BATCH * FEAT)] = acc[r];
  }
}

// ---------------------------------------------------------------------------
// Stage 2 reduce + tanh: out[k][b][f] = tanh(sum over NCHUNKS partials).
// ---------------------------------------------------------------------------
__global__ __launch_bounds__(256)
void k_out_reduce(const float* __restrict__ ws, float* __restrict__ out) {
  const size_t i = (size_t)blockIdx.x * 256 + threadIdx.x;  // 8*64*128 = 65536
  float s = 0.f;
  #pragma unroll
  for (int c = 0; c < NCHUNKS; ++c)
    s += ws[WS2_OFF + (size_t)c * (N_FILT * BATCH * FEAT) + i];
  out[i] = tanhf(s);
}

// ---------------------------------------------------------------------------
// Launch
// ---------------------------------------------------------------------------
extern "C" void kernel_launch(void* const* d_in, const int* in_sizes, int n_in,
                              void* d_out, int out_size, void* d_ws, size_t ws_size,
                              hipStream_t stream) {
  const float* x = (const float*)d_in[0];   // [64, 8192, 128]
  const float* L = (const float*)d_in[1];   // [8192, 8192]
  const float* W = (const float*)d_in[2];   // [8, 8192]
  float* out = (float*)d_out;               // [8, 64, 128]
  float* ws  = (float*)d_ws;                // ~7.4 MB used

  // Prologue: zero-padded A matrix for stage 1
  k_pad_W<<<512, 256, 0, stream>>>(W, ws);
  // Stage 1: 4096 waves (512 N-tiles x 8 K-chunks), 4 waves/block
  k_wL_partial<<<1024, 128, 0, stream>>>(L, ws);
  // Reduce partial wL -> wLpad[16][8192]
  k_wL_reduce<<<512, 256, 0, stream>>>(ws);
  // Stage 2: 4096 waves (8 n-chunks x 64 batches x 8 f-tiles)
  k_out_partial<<<1024, 128, 0, stream>>>(x, ws);
  // Reduce + tanh -> out
  k_out_reduce<<<256, 256, 0, stream>>>(ws, out);
}